// IF_18622978195596
// MI455X (gfx1250) — compile-verified
//
#include <hip/hip_runtime.h>

// Integrate-and-fire SNN forward, T=4 timesteps.
// x: [T*B, C, H, W] fp32; out: spikes, same shape. mem0 = 0.5*thresh.
// Memory-bound: 268 MB total traffic -> ~11.5 us floor at 23.3 TB/s.

typedef float v4f __attribute__((ext_vector_type(4)));
typedef int   v4i __attribute__((ext_vector_type(4)));

// Pointer types matching the async builtin's signature (revealed by the
// round-1 diagnostic): AS(1) int4* source, AS(3) int4* LDS destination.
typedef __attribute__((address_space(1))) v4i* gas_b128_t;
typedef __attribute__((address_space(3))) v4i* las_b128_t;

#define T_STEPS 4
#define BLOCK 256

// CDNA5 async global->LDS path (per-lane LDS-DMA, tracked by ASYNCcnt).
#if __has_builtin(__builtin_amdgcn_global_load_async_to_lds_b128) && \
    __has_builtin(__builtin_amdgcn_s_wait_asynccnt)
#define USE_ASYNC_LDS 1
#else
#define USE_ASYNC_LDS 0
#endif

__global__ __launch_bounds__(BLOCK) void if_stream_kernel(
    const v4f* __restrict__ x,      // T_STEPS * n4 float4s
    const float* __restrict__ threshp,
    v4f* __restrict__ out,          // T_STEPS * n4 float4s
    unsigned int n4)                // float4 elements per timestep
{
    const float th = *threshp;                       // uniform scalar load
    const unsigned int i = blockIdx.x * BLOCK + threadIdx.x;
    if (i >= n4) return;

    v4f xv[T_STEPS];

#if USE_ASYNC_LDS
    // Stage the 4 timestep slices through LDS with the CDNA5 async engine.
    // Each lane's 4 loads are issued back-to-back (no dest VGPRs held while
    // in flight), then a single s_wait_asynccnt 0 covers all of them. Each
    // lane reads back only the slots it requested -> no barrier required.
    __shared__ v4f tile[T_STEPS][BLOCK];
    #pragma unroll
    for (int t = 0; t < T_STEPS; ++t) {
        __builtin_amdgcn_global_load_async_to_lds_b128(
            (gas_b128_t)(x + (size_t)t * n4 + i),
            (las_b128_t)&tile[t][threadIdx.x],
            /*imm offset*/0, /*cpol*/0);
    }
    __builtin_amdgcn_s_wait_asynccnt(0);
    #pragma unroll
    for (int t = 0; t < T_STEPS; ++t) {
        xv[t] = tile[t][threadIdx.x];                // ds_load_b128
    }
#else
    // Direct streaming path: 4 independent non-temporal B128 loads up front
    // for maximum memory-level parallelism; NT since every byte is read once
    // and the 268 MB stream exceeds the 192 MB L2.
    #pragma unroll
    for (int t = 0; t < T_STEPS; ++t) {
        xv[t] = __builtin_nontemporal_load(x + (size_t)t * n4 + i);
    }
#endif

    const float h = 0.5f * th;
    v4f mem = {h, h, h, h};

    #pragma unroll
    for (int t = 0; t < T_STEPS; ++t) {
        mem += xv[t];
        v4f spike;
        spike.x = (mem.x >= th) ? th : 0.0f;
        spike.y = (mem.y >= th) ? th : 0.0f;
        spike.z = (mem.z >= th) ? th : 0.0f;
        spike.w = (mem.w >= th) ? th : 0.0f;
        mem -= spike;
        // NT store: written once, never re-read by this dispatch.
        __builtin_nontemporal_store(spike, out + (size_t)t * n4 + i);
    }
}

extern "C" void kernel_launch(void* const* d_in, const int* in_sizes, int n_in,
                              void* d_out, int out_size, void* d_ws, size_t ws_size,
                              hipStream_t stream) {
    (void)n_in; (void)d_ws; (void)ws_size; (void)out_size;
    const v4f*   x      = (const v4f*)d_in[0];
    const float* thresh = (const float*)d_in[1];
    // d_in[2] is T on device; it is a compile-time constant (4) of the
    // reference, and reading it on host would break graph capture.

    const long long total = (long long)in_sizes[0];     // T*B*C*H*W floats
    const long long perT  = total / T_STEPS;            // floats per timestep
    const unsigned int n4 = (unsigned int)(perT / 4);   // float4s per timestep

    dim3 block(BLOCK);
    dim3 grid((n4 + BLOCK - 1) / BLOCK);
    if_stream_kernel<<<grid, block, 0, stream>>>(x, thresh, (v4f*)d_out, n4);
}